// DeployModule_76871324663865
// MI455X (gfx1250) — compile-verified
//
#include <hip/hip_runtime.h>
#include <hip/hip_bf16.h>
#include <math.h>
#include <stdint.h>

// ---------------- problem constants ----------------
#define NB      32
#define NA      8400
#define NCH     85
#define NCLS    80
#define MAXDET  300
#define CONF_THRE 0.2f
#define NMS_THRE  0.45f

// kernel 1 tiling
#define ANCH_PER_BLK 128
#define BLK1         128
// top-k two-stage sort
#define NSEG   4          // segments per batch
#define SEGLEN 2100       // NA / NSEG exactly
#define SEGN   4096       // padded per-segment sort size
#define KEEPK  512        // keys kept per segment (>= MAXDET)
#define MRGN   2048       // NSEG * KEEPK, stage-B sort size
#define BLKS   1024
// nms
#define BLK3   512
#define NW     10         // ceil(300/32) mask words

// output offsets (floats)
#define O_BOX    0                        // 32*300*4 = 38400
#define O_INZONE (NB*MAXDET*4)            // 38400
#define O_SCORE  (O_INZONE + NB*MAXDET)   // 48000
#define O_CLS    (O_SCORE  + NB*MAXDET)   // 57600
#define O_CTR    (O_CLS    + NB*MAXDET)   // 67200
#define O_KEEP   (O_CTR    + NB*MAXDET*2) // 86400

#define AS1 __attribute__((address_space(1)))
#define AS3 __attribute__((address_space(3)))

typedef __attribute__((__vector_size__(16))) int v4i;
typedef unsigned long long u64;

#if defined(__has_builtin)
#  if __has_builtin(__builtin_amdgcn_global_load_async_to_lds_b128)
#    define HAVE_ASYNC_B128 1
#  endif
#  if __has_builtin(__builtin_amdgcn_s_wait_asynccnt)
#    define HAVE_WAIT_ASYNC 1
#  endif
#endif

// CDNA5 async global->LDS copy of one 16B chunk (ASYNCcnt-tracked).
__device__ __forceinline__ void async_copy_b128(const float* g, float* l) {
#ifdef HAVE_ASYNC_B128
  __builtin_amdgcn_global_load_async_to_lds_b128(
      (AS1 v4i*)g, (AS3 v4i*)l, /*offset=*/0, /*cpol=*/0);
#else
  unsigned lds_off = (unsigned)(uintptr_t)(AS3 void*)l;
  unsigned long long ga = (unsigned long long)(uintptr_t)g;
  asm volatile("global_load_async_to_lds_b128 %0, %1, off"
               :: "v"(lds_off), "v"(ga) : "memory");
#endif
}

__device__ __forceinline__ void wait_async_0() {
#ifdef HAVE_WAIT_ASYNC
  __builtin_amdgcn_s_wait_asynccnt(0);
#else
  asm volatile("s_wait_asynccnt 0" ::: "memory");
#endif
}

// order-preserving float->u32 map (descending u32 order == descending float)
__device__ __forceinline__ unsigned fkey(float v) {
  unsigned u = __float_as_uint(v);
  return (u & 0x80000000u) ? ~u : (u | 0x80000000u);
}
__device__ __forceinline__ float funkey(unsigned u) {
  return (u & 0x80000000u) ? __uint_as_float(u & 0x7FFFFFFFu)
                           : __uint_as_float(~u);
}

// in-LDS descending bitonic sort of N u64 keys, NT threads
template <int N, int NT>
__device__ __forceinline__ void bitonic_desc(u64* keys, int tid) {
  for (unsigned k = 2; k <= (unsigned)N; k <<= 1) {
    for (unsigned j = k >> 1; j > 0; j >>= 1) {
      __syncthreads();
      for (unsigned i = tid; i < (unsigned)N; i += NT) {
        const unsigned ix = i ^ j;
        if (ix > i) {
          const u64 a = keys[i], c = keys[ix];
          const bool descBlock = ((i & k) == 0);
          if (descBlock ? (a < c) : (a > c)) { keys[i] = c; keys[ix] = a; }
        }
      }
    }
  }
  __syncthreads();
}

// ---------------------------------------------------------------------------
// Kernel 1: stream prediction through LDS (async b128), decode boxes + score.
// payload[anchor] = {x0,y0,x1,y1, obj, class_conf, cls, nms_score}
// ---------------------------------------------------------------------------
__global__ void score_kernel(const float* __restrict__ pred,
                             float* __restrict__ payload,
                             float* __restrict__ scores) {
  // 16B alignment required: async b128 writes LDS with B128 native alignment.
  __shared__ __align__(16) float tile[ANCH_PER_BLK * NCH];   // 43520 B
  const int b  = blockIdx.y;
  const int a0 = blockIdx.x * ANCH_PER_BLK;
  int rows = NA - a0;
  if (rows > ANCH_PER_BLK) rows = ANCH_PER_BLK;
  const int tid = threadIdx.x;

  // tile base 16B aligned; rows is 128 or 80 (both %4==0) so rows*85*4 is a
  // whole number of 16B chunks; global base is 16B aligned (a0%128==0).
  const float* gbase = pred + ((size_t)b * NA + a0) * NCH;
  const int chunks = (rows * NCH) >> 2;
  for (int c = tid; c < chunks; c += BLK1)
    async_copy_b128(gbase + (c << 2), tile + (c << 2));
  wait_async_0();
  __syncthreads();

  if (tid < rows) {
    const float* r = tile + tid * NCH;         // gcd(85,64)=1 -> conflict-free
    const float cx = r[0], cy = r[1], w = r[2], h = r[3];
    const float obj = r[4];
    float best = r[5];
    int   bi   = 0;
#pragma unroll 8
    for (int k = 1; k < NCLS; ++k) {
      const float v = r[5 + k];
      if (v > best) { best = v; bi = k; }      // first-max like jnp.argmax
    }
    float s = obj * best;
    if (!(s >= CONF_THRE)) s = -INFINITY;
    const float x0 = cx - 0.5f * w, y0 = cy - 0.5f * h;
    const float x1 = cx + 0.5f * w, y1 = cy + 0.5f * h;
    const size_t aidx = (size_t)b * NA + a0 + tid;
    float4* p = (float4*)(payload + aidx * 8);
    p[0] = make_float4(x0, y0, x1, y1);
    p[1] = make_float4(obj, best, (float)bi, s);
    scores[aidx] = s;
  }
}

// ---------------------------------------------------------------------------
// Kernel 2a: per-segment sort (4 segments x 32 batches = 128 blocks).
// key = fkey(score)<<32 | ~global_idx  (padding: low 32 bits = 0, sorts last)
// ---------------------------------------------------------------------------
__global__ void segsort_kernel(const float* __restrict__ scores,
                               u64* __restrict__ topkeys) {
  __shared__ u64 keys[SEGN];                   // 32 KB
  const int b = blockIdx.y, seg = blockIdx.x, tid = threadIdx.x;
  const int base = seg * SEGLEN;
  const float* s = scores + (size_t)b * NA;

  for (int i = tid; i < SEGN; i += BLKS) {
    u64 key;
    if (i < SEGLEN) {
      const int gi = base + i;
      key = ((u64)fkey(s[gi]) << 32) | (unsigned)(~(unsigned)gi);
    } else {
      key = ((u64)0x007FFFFFu << 32);          // fkey(-inf), idx field = ~0
    }
    keys[i] = key;
  }
  bitonic_desc<SEGN, BLKS>(keys, tid);
  if (tid < KEEPK)
    topkeys[((size_t)b * NSEG + seg) * KEEPK + tid] = keys[tid];
}

// ---------------------------------------------------------------------------
// Kernel 2b: merge 4x512 per batch -> exact sorted top-300.
// ---------------------------------------------------------------------------
__global__ void merge_kernel(const u64* __restrict__ topkeys,
                             float* __restrict__ topscore,
                             int*   __restrict__ topidx) {
  __shared__ u64 keys[MRGN];                   // 16 KB
  const int b = blockIdx.x, tid = threadIdx.x;
  for (int i = tid; i < MRGN; i += BLKS)
    keys[i] = topkeys[(size_t)b * MRGN + i];
  bitonic_desc<MRGN, BLKS>(keys, tid);
  if (tid < MAXDET) {
    const u64 key = keys[tid];
    const float v = funkey((unsigned)(key >> 32));
    int idx = (int)(~(unsigned)key);
    if (idx < 0 || idx >= NA) idx = 0;         // padding safety
    topscore[b * MAXDET + tid] = v;
    topidx[b * MAXDET + tid]   = idx;
  }
}

// ---------------------------------------------------------------------------
// Kernel 3: suppression-matrix NMS (no per-step barriers), zone test, outputs.
// ---------------------------------------------------------------------------
__global__ void nms_kernel(const float* __restrict__ payload,
                           const float* __restrict__ topscore,
                           const int*   __restrict__ topidx,
                           const float* __restrict__ zone,
                           float* __restrict__ out) {
  __shared__ __align__(16) float lx0[MAXDET], ly0[MAXDET], lx1[MAXDET],
                                 ly1[MAXDET], larea[MAXDET];
  __shared__ int      lkeep[MAXDET];
  __shared__ unsigned rows[MAXDET * NW];       // sup bitmask rows, 12 KB
  __shared__ unsigned maskw[NW];
  __shared__ float zx[8], zy[8];
  const int b = blockIdx.x, j = threadIdx.x;

  if (j < 8)       zx[j]     = zone[j * 2 + 0];
  else if (j < 16) zy[j - 8] = zone[(j - 8) * 2 + 1];

  float x0 = 0.f, y0 = 0.f, x1 = 0.f, y1 = 0.f;
  float obj = 0.f, conf = 0.f, clsf = 0.f;
  if (j < MAXDET) {
    const float sscore = topscore[b * MAXDET + j];
    const int   idx    = topidx[b * MAXDET + j];
    const float4* p = (const float4*)(payload + ((size_t)b * NA + idx) * 8);
    const float4 p0 = p[0], p1 = p[1];
    x0 = p0.x; y0 = p0.y; x1 = p0.z; y1 = p0.w;
    obj = p1.x; conf = p1.y; clsf = p1.z;
    lx0[j] = x0; ly0[j] = y0; lx1[j] = x1; ly1[j] = y1;
    larea[j] = fmaxf(x1 - x0, 0.f) * fmaxf(y1 - y0, 0.f);
    lkeep[j] = (sscore > -INFINITY) ? 1 : 0;   // jnp.isfinite(s)
  }
  __syncthreads();

  // initial keep mask words
  if (j < NW) {
    unsigned wbits = 0;
    for (int t = 0; t < 32; ++t) {
      const int q = j * 32 + t;
      if (q < MAXDET && lkeep[q]) wbits |= (1u << t);
    }
    maskw[j] = wbits;
  }
  // suppression rows: task (i, word) -> 32 IoU tests
  for (int t = j; t < MAXDET * NW; t += BLK3) {
    const int i = t / NW, w = t % NW;
    const float ax0 = lx0[i], ay0 = ly0[i], ax1 = lx1[i], ay1 = ly1[i];
    const float aarea = larea[i];
    unsigned bits = 0;
    const int j0 = w * 32;
#pragma unroll 8
    for (int q = 0; q < 32; ++q) {
      const int jj = j0 + q;
      if (jj < MAXDET && jj > i) {
        const float iw = fmaxf(fminf(lx1[jj], ax1) - fmaxf(lx0[jj], ax0), 0.f);
        const float ih = fmaxf(fminf(ly1[jj], ay1) - fmaxf(ly0[jj], ay0), 0.f);
        const float inter = iw * ih;
        const float iou = inter / (aarea + larea[jj] - inter + 1e-9f);
        if (iou > NMS_THRE) bits |= (1u << q);
      }
    }
    rows[i * NW + w] = bits;
  }
  __syncthreads();

  // serial greedy walk: wave 0, lanes 0..9 each own one mask word.
  if (j < NW) {
    unsigned m = maskw[j];
    for (int i = 0; i < MAXDET; ++i) {
      const unsigned mo = __shfl(m, i >> 5);   // word owning bit i
      if ((mo >> (i & 31)) & 1u) m &= ~rows[i * NW + j];
    }
    maskw[j] = m;
  }
  __syncthreads();

  if (j < MAXDET) {
    const bool keep = (maskw[j >> 5] >> (j & 31)) & 1u;
    const float cx = 0.5f * (x0 + x1), cy = 0.5f * (y0 + y1);
    int cnt = 0;
#pragma unroll
    for (int v = 0; v < 8; ++v) {
      const int vp = (v + 7) & 7;              // roll(zone, 1)
      const float xi = zx[v],  yi = zy[v];
      const float xj = zx[vp], yj = zy[vp];
      const bool gyi = yi > cy, gyj = yj > cy;
      const float gx = (xj - xi) * (cy - yi) / (yj - yi) + xi;
      if ((gyi != gyj) && (gx > cx)) cnt++;
    }
    const bool inzone = ((cnt & 1) != 0) && keep;
    const size_t r = (size_t)b * MAXDET + j;
    float4* bo = (float4*)(out + O_BOX + r * 4);
    *bo = make_float4(y0, x0, y1, x1);         // boxes_yxyx
    out[O_INZONE + r] = inzone ? 1.f : 0.f;
    out[O_SCORE  + r] = fmaxf(obj, conf);
    out[O_CLS    + r] = clsf;
    float* ctr = out + O_CTR + r * 2;
    ctr[0] = cy; ctr[1] = cx;                  // centers_yx
    out[O_KEEP + r] = keep ? 1.f : 0.f;
  }
}

// ---------------------------------------------------------------------------
extern "C" void kernel_launch(void* const* d_in, const int* in_sizes, int n_in,
                              void* d_out, int out_size, void* d_ws, size_t ws_size,
                              hipStream_t stream) {
  const float* pred = (const float*)d_in[0];   // (32, 8400, 85)
  const float* zone = (const float*)d_in[1];   // (8, 2)
  float* out = (float*)d_out;

  // workspace layout: payload | scores | topkeys(u64) | topscore | topidx
  float* payload  = (float*)d_ws;                          // 32*8400*8 f
  float* scores   = payload + (size_t)NB * NA * 8;         // 32*8400 f
  u64*   topkeys  = (u64*)(scores + (size_t)NB * NA);      // 32*4*512 u64 (8B-aligned)
  float* topscore = (float*)(topkeys + (size_t)NB * NSEG * KEEPK); // 32*300 f
  int*   topidx   = (int*)(topscore + NB * MAXDET);        // 32*300 i

  dim3 g1((NA + ANCH_PER_BLK - 1) / ANCH_PER_BLK, NB);
  score_kernel<<<g1, BLK1, 0, stream>>>(pred, payload, scores);
  dim3 g2(NSEG, NB);
  segsort_kernel<<<g2, BLKS, 0, stream>>>(scores, topkeys);
  merge_kernel<<<NB, BLKS, 0, stream>>>(topkeys, topscore, topidx);
  nms_kernel<<<NB, BLK3, 0, stream>>>(payload, topscore, topidx, zone, out);
}